// Self_Attention_v1_43817256353888
// MI455X (gfx1250) — compile-verified
//
#include <hip/hip_runtime.h>
#include <math.h>

typedef float v2f __attribute__((ext_vector_type(2)));
typedef float v8f __attribute__((ext_vector_type(8)));

constexpr int Bb = 8, Cc = 64, Nn = 4096, Ee = 8;

// D = A(16x4, f32) * B(4x16, f32) + C(16x16, f32)
#define WMMA_F32_16X16X4(A, B, C) \
  __builtin_amdgcn_wmma_f32_16x16x4_f32(false, (A), false, (B), (short)0, (C), false, false)

// lane l <-> lane l^16 exchange via v_permlanex16_b32 (identity lane selects).
// Pure VALU: avoids ds_bpermute + s_wait_dscnt on the critical path.
__device__ __forceinline__ float swap_half(float v) {
  int i = __builtin_bit_cast(int, v);
  int r = __builtin_amdgcn_permlanex16(i, i, 0x76543210u, 0xfedcba98u, false, false);
  return __builtin_bit_cast(float, r);
}

// ---------------------------------------------------------------------------
// Kernel 1: f/g/h = W{k,q,v} @ x + b   -> [B, E, N] each
// ---------------------------------------------------------------------------
__global__ __launch_bounds__(256) void proj_kernel(
    const float* __restrict__ x,
    const float* __restrict__ Wk, const float* __restrict__ bk,
    const float* __restrict__ Wq, const float* __restrict__ bq,
    const float* __restrict__ Wv, const float* __restrict__ bv,
    float* __restrict__ F, float* __restrict__ G, float* __restrict__ H) {
  __shared__ float sW[3 * Ee * Cc];
  __shared__ float sb[3 * Ee];
  for (int i = threadIdx.x; i < Ee * Cc; i += blockDim.x) {
    sW[i]               = Wk[i];
    sW[Ee * Cc + i]     = Wq[i];
    sW[2 * Ee * Cc + i] = Wv[i];
  }
  if (threadIdx.x < Ee) {
    sb[threadIdx.x]          = bk[threadIdx.x];
    sb[Ee + threadIdx.x]     = bq[threadIdx.x];
    sb[2 * Ee + threadIdx.x] = bv[threadIdx.x];
  }
  __syncthreads();

  const int gid = blockIdx.x * blockDim.x + threadIdx.x;   // 0 .. B*N-1
  const int b = gid >> 12;           // N == 4096
  const int n = gid & (Nn - 1);
  const float* xp = x + (size_t)b * Cc * Nn + n;

  float aF[Ee], aG[Ee], aH[Ee];
#pragma unroll
  for (int e = 0; e < Ee; ++e) { aF[e] = sb[e]; aG[e] = sb[Ee + e]; aH[e] = sb[2 * Ee + e]; }

  for (int c = 0; c < Cc; ++c) {
    const float xc = xp[(size_t)c * Nn];
#pragma unroll
    for (int e = 0; e < Ee; ++e) {
      aF[e] += sW[e * Cc + c] * xc;
      aG[e] += sW[Ee * Cc + e * Cc + c] * xc;
      aH[e] += sW[2 * Ee * Cc + e * Cc + c] * xc;
    }
  }
#pragma unroll
  for (int e = 0; e < Ee; ++e) {
    const size_t o = ((size_t)b * Ee + e) * Nn + n;
    F[o] = aF[e]; G[o] = aG[e]; H[o] = aH[e];
  }
}

// ---------------------------------------------------------------------------
// Kernel 2: fused flash-style attention, softmax over key axis.
// One wave (32 lanes) handles one (batch, 16-column query tile).
// S-tile and PV via V_WMMA_F32_16X16X4_F32; cross-half exchanges via
// v_permlanex16 instead of LDS bpermute.
// ---------------------------------------------------------------------------
__global__ __launch_bounds__(128) void attn_kernel(
    const float* __restrict__ F, const float* __restrict__ G,
    const float* __restrict__ H, float* __restrict__ V) {
  const int wave = threadIdx.x >> 5;
  const int lane = threadIdx.x & 31;
  const int tile = blockIdx.x * 4 + wave;   // 0 .. B*(N/16)-1
  const int b  = tile >> 8;                 // 256 m-tiles per batch
  const int m0 = (tile & 255) << 4;
  const int lm = lane & 15;
  const bool hi = lane >= 16;

  const float* Fb = F + (size_t)b * Ee * Nn;
  const float* Gb = G + (size_t)b * Ee * Nn;
  const float* Hb = H + (size_t)b * Ee * Nn;

  const int m  = m0 + lm;
  const int eo = hi ? 2 : 0;                // e-offset for the lane's half

  // B operand for S = F^T G (K = e chunks of 4), loaded once
  v2f bq0 = { Gb[(size_t)(eo + 0) * Nn + m], Gb[(size_t)(eo + 1) * Nn + m] };
  v2f bq1 = { Gb[(size_t)(eo + 4) * Nn + m], Gb[(size_t)(eo + 5) * Nn + m] };

  // A-row (e) for the PV WMMA; rows 8..15 are zero padding
  const int  eA = lane & 15;
  const bool realE = eA < Ee;
  const float* Hrow = Hb + (size_t)(realE ? eA : 0) * Nn;
  const int hOff = hi ? 2 : 0;

  float runm = -__builtin_inff();
  float runs = 0.f;
  v8f acc = {0.f, 0.f, 0.f, 0.f, 0.f, 0.f, 0.f, 0.f};

  for (int n0 = 0; n0 < Nn; n0 += 16) {
    const int nI = n0 + lm;
    __builtin_prefetch(&Fb[(size_t)(eo + 0) * Nn + nI + 512], 0, 0);
    __builtin_prefetch(&Hrow[n0 + 512], 0, 0);

    // ---- S-tile: S[n0+r, m0+c] = sum_e F[e,n] G[e,m] (two K=4 chunks) ----
    v2f a0 = { Fb[(size_t)(eo + 0) * Nn + nI], Fb[(size_t)(eo + 1) * Nn + nI] };
    v2f a1 = { Fb[(size_t)(eo + 4) * Nn + nI], Fb[(size_t)(eo + 5) * Nn + nI] };
    v8f s = {0.f, 0.f, 0.f, 0.f, 0.f, 0.f, 0.f, 0.f};
    s = WMMA_F32_16X16X4(a0, bq0, s);
    s = WMMA_F32_16X16X4(a1, bq1, s);
    // lane l<16 holds S rows 0..7 of column m0+l; lane l+16 holds rows 8..15

    // ---- online softmax over key axis (column-wise) ----
    float tmax = s[0];
#pragma unroll
    for (int r = 1; r < 8; ++r) tmax = fmaxf(tmax, s[r]);
    tmax = fmaxf(tmax, swap_half(tmax));
    const float newm = fmaxf(runm, tmax);

    float p[8];
    float tsum = 0.f;
#pragma unroll
    for (int r = 0; r < 8; ++r) { p[r] = __expf(s[r] - newm); tsum += p[r]; }
    tsum += swap_half(tsum);

    const float rescale = __expf(runm - newm);   // 0 on first iter (runm = -inf)
    runs = runs * rescale + tsum;
    runm = newm;
#pragma unroll
    for (int r = 0; r < 8; ++r) acc[r] *= rescale;

    // partner-half P values for B-operand construction
    float q[8];
#pragma unroll
    for (int r = 0; r < 8; ++r) q[r] = swap_half(p[r]);

    // ---- PV: acc[e, m] += H[e, n-chunk] * P[n-chunk, m], 4 K=4 chunks ----
#pragma unroll
    for (int j = 0; j < 4; ++j) {
      float h0 = 0.f, h1 = 0.f;
      if (realE) {
        h0 = Hrow[n0 + 4 * j + hOff];
        h1 = Hrow[n0 + 4 * j + hOff + 1];
      }
      v2f hA = { h0, h1 };
      const int base = (j < 2) ? 4 * j : 4 * (j - 2);
      v2f bP;
      bP.x = hi ? ((j < 2) ? q[base + 2] : p[base + 2])
                : ((j < 2) ? p[base + 0] : q[base + 0]);
      bP.y = hi ? ((j < 2) ? q[base + 3] : p[base + 3])
                : ((j < 2) ? p[base + 1] : q[base + 1]);
      acc = WMMA_F32_16X16X4(hA, bP, acc);
    }
  }

  const float inv = 1.0f / runs;
  if (!hi) {   // lane l<16 holds V[e=0..7][m0+l]; padded rows are zero
#pragma unroll
    for (int r = 0; r < 8; ++r)
      V[((size_t)b * Ee + r) * Nn + m] = acc[r] * inv;
  }
}

// ---------------------------------------------------------------------------
// Kernel 3: o = Wo @ v + bo ; y = gamma*o + x ; emit (y, o, gamma)
// Streaming outputs use non-temporal stores (never re-read).
// ---------------------------------------------------------------------------
__global__ __launch_bounds__(256) void out_kernel(
    const float* __restrict__ x,
    const float* __restrict__ Wo, const float* __restrict__ bo,
    const float* __restrict__ gamma, const float* __restrict__ V,
    float* __restrict__ y, float* __restrict__ o_out, float* __restrict__ g_out) {
  __shared__ float sWo[Cc * Ee];
  __shared__ float sbo[Cc];
  for (int i = threadIdx.x; i < Cc * Ee; i += blockDim.x) sWo[i] = Wo[i];
  if (threadIdx.x < Cc) sbo[threadIdx.x] = bo[threadIdx.x];
  __syncthreads();

  const int gid = blockIdx.x * blockDim.x + threadIdx.x;
  const int b = gid >> 12;
  const int n = gid & (Nn - 1);

  float v[Ee];
#pragma unroll
  for (int e = 0; e < Ee; ++e) v[e] = V[((size_t)b * Ee + e) * Nn + n];

  const float g = gamma[0];
  for (int c = 0; c < Cc; ++c) {
    float o = sbo[c];
#pragma unroll
    for (int e = 0; e < Ee; ++e) o += sWo[c * Ee + e] * v[e];
    const size_t idx = ((size_t)b * Cc + c) * Nn + n;
    __builtin_nontemporal_store(o, &o_out[idx]);
    __builtin_nontemporal_store(g * o + x[idx], &y[idx]);
  }
  if (gid == 0) g_out[0] = g;
}

// ---------------------------------------------------------------------------
extern "C" void kernel_launch(void* const* d_in, const int* in_sizes, int n_in,
                              void* d_out, int out_size, void* d_ws, size_t ws_size,
                              hipStream_t stream) {
  const float* x     = (const float*)d_in[0];
  const float* Wk    = (const float*)d_in[1];
  const float* bk    = (const float*)d_in[2];
  const float* Wq    = (const float*)d_in[3];
  const float* bq    = (const float*)d_in[4];
  const float* Wv    = (const float*)d_in[5];
  const float* bv    = (const float*)d_in[6];
  const float* Wo    = (const float*)d_in[7];
  const float* bo    = (const float*)d_in[8];
  const float* gamma = (const float*)d_in[9];

  float* out   = (float*)d_out;
  float* y     = out;                               // [B,C,H,W]
  float* o     = out + (size_t)Bb * Cc * Nn;        // [B,C,H,W]
  float* g_out = out + 2 * (size_t)Bb * Cc * Nn;    // scalar

  float* F = (float*)d_ws;                          // [B,E,N]
  float* G = F + (size_t)Bb * Ee * Nn;
  float* H = G + (size_t)Bb * Ee * Nn;
  float* V = H + (size_t)Bb * Ee * Nn;

  proj_kernel<<<(Bb * Nn) / 256, 256, 0, stream>>>(x, Wk, bk, Wq, bq, Wv, bv, F, G, H);
  attn_kernel<<<(Bb * Nn / 16) / 4, 128, 0, stream>>>(F, G, H, V);
  out_kernel<<<(Bb * Nn) / 256, 256, 0, stream>>>(x, Wo, bo, gamma, V, y, o, g_out);
}